// SAGEConvolutionLin_72911364817008
// MI455X (gfx1250) — compile-verified
//
#include <hip/hip_runtime.h>
#include <math.h>

// GraphSAGE 2-layer inference for MI455X (gfx1250, wave32).
// Scratch layout (floats): deg_inv[N] | bufA[N*128] | bufB[N*128]  (~103 MB)
// Whole working set is L2-resident (192 MB L2), so scatter/gather run out of L2.

#define NN   100000   // nodes
#define NE   1600000  // edges
#define DH   128      // F_IN == HID
#define NC   40       // classes

typedef float v2f __attribute__((ext_vector_type(2)));
typedef float v8f __attribute__((ext_vector_type(8)));

// ---------------------------------------------------------------- utilities
__global__ void zero_kernel(float* __restrict__ p, int n) {
  int i = blockIdx.x * blockDim.x + threadIdx.x;
  if (i < n) p[i] = 0.0f;
}

__global__ void deg_count_kernel(const int* __restrict__ dst,
                                 float* __restrict__ deg, int n_edges) {
  int e = blockIdx.x * blockDim.x + threadIdx.x;
  if (e < n_edges) atomicAdd(&deg[dst[e]], 1.0f);
}

__global__ void deg_invert_kernel(float* __restrict__ deg, int n) {
  int i = blockIdx.x * blockDim.x + threadIdx.x;
  if (i < n) {
    float d = deg[i];
    deg[i] = (d > 0.0f) ? (1.0f / d) : 0.0f;
  }
}

// ------------------------------------------------------- edge gather/scatter
// 32 lanes per edge: lane l handles features 4l..4l+3 (float4 gather, 4 f32
// atomics). Consecutive lanes cover one 512B source row -> fully coalesced.
__global__ __launch_bounds__(256) void scatter_add_kernel(
    const float* __restrict__ X, const int* __restrict__ src,
    const int* __restrict__ dst, float* __restrict__ agg, int n_edges) {
  long long tid = (long long)blockIdx.x * blockDim.x + threadIdx.x;
  int e    = (int)(tid >> 5);
  int lane = (int)(tid & 31);
  if (e >= n_edges) return;
  int s = src[e];
  int d = dst[e];
  const float4 v = *(const float4*)(X + (size_t)s * DH + lane * 4);
  float* o = agg + (size_t)d * DH + lane * 4;
  atomicAdd(o + 0, v.x);
  atomicAdd(o + 1, v.y);
  atomicAdd(o + 2, v.z);
  atomicAdd(o + 3, v.w);
}

// --------------------------------------------------------------- WMMA GEMM
// D[16-row stripe] = (dinv .* A1) @ W1 + A2 @ W2 + bias   (optional ReLU)
// 256 threads = 8 waves; wave w computes column tile n0 = 16*w.
// fp32 WMMA 16x16x4: A-tile 16x4 (v2f/lane), B-tile 4x16 (v2f/lane), C v8f.
// All A reads complete (consumed by WMMA) before the barrier, so D may alias
// A1/A2 in-place (each block exclusively owns its 16 rows).
__global__ __launch_bounds__(256) void gemm_dual_wmma(
    const float* __restrict__ A1, const float* __restrict__ dinv,
    const float* __restrict__ W1,
    const float* __restrict__ A2, const float* __restrict__ W2,
    const float* __restrict__ bias, float* __restrict__ D, int relu) {
  const int l     = threadIdx.x & 31;        // lane in wave
  const int wave  = threadIdx.x >> 5;        // 0..7 -> column tile
  const int m0    = blockIdx.x * 16;
  const int n0    = wave * 16;
  const int mrow  = m0 + (l & 15);           // A-fragment row (lane-invariant in k)
  const int khalf = (l >> 4) << 1;           // 0 (lanes 0-15) or 2 (lanes 16-31)
  const int ncol  = n0 + (l & 15);           // B/C fragment column
  const float s   = dinv[mrow];              // mean-aggregation scale on A1

  const float* a1p = A1 + (size_t)mrow * DH + khalf;
  const float* a2p = A2 + (size_t)mrow * DH + khalf;

  v8f acc1 = {0.f, 0.f, 0.f, 0.f, 0.f, 0.f, 0.f, 0.f};
  v8f acc2 = {0.f, 0.f, 0.f, 0.f, 0.f, 0.f, 0.f, 0.f};

#pragma unroll 4
  for (int k0 = 0; k0 < DH; k0 += 4) {
    v2f a1 = *(const v2f*)(a1p + k0);
    a1.x *= s;
    a1.y *= s;
    v2f b1;
    b1.x = W1[(k0 + khalf) * DH + ncol];
    b1.y = W1[(k0 + khalf + 1) * DH + ncol];
    acc1 = __builtin_amdgcn_wmma_f32_16x16x4_f32(
        /*neg_a=*/false, a1, /*neg_b=*/false, b1,
        /*c_mod=*/(short)0, acc1, /*reuse_a=*/false, /*reuse_b=*/false);

    v2f a2 = *(const v2f*)(a2p + k0);
    v2f b2;
    b2.x = W2[(k0 + khalf) * DH + ncol];
    b2.y = W2[(k0 + khalf + 1) * DH + ncol];
    acc2 = __builtin_amdgcn_wmma_f32_16x16x4_f32(
        false, a2, false, b2, (short)0, acc2, false, false);
  }

  // Guarantee all A-tile loads have landed in registers, then barrier so no
  // wave in this block stores over rows another wave is still reading
  // (required for the in-place layer-2 call where D == A1).
  asm volatile("s_wait_loadcnt 0" ::: "memory");
  __syncthreads();

  const float bn = bias[ncol];
  const int mbase = m0 + ((l >> 4) << 3);    // C layout: VGPR r -> M = r or r+8
#pragma unroll
  for (int r = 0; r < 8; ++r) {
    float v = acc1[r] + acc2[r] + bn;
    if (relu) v = fmaxf(v, 0.0f);
    D[(size_t)(mbase + r) * DH + ncol] = v;
  }
}

// ----------------------------------------------- classifier + log_softmax
__global__ __launch_bounds__(64) void out_logsoftmax_kernel(
    const float* __restrict__ H, const float* __restrict__ Wout,
    const float* __restrict__ bout, float* __restrict__ out) {
  __shared__ float logits[NC];
  __shared__ float red[2];
  const int node = blockIdx.x;
  const int t = threadIdx.x;
  if (t < NC) {
    float acc = bout[t];
    const float* h = H + (size_t)node * DH;
#pragma unroll 4
    for (int k = 0; k < DH; ++k)
      acc += h[k] * Wout[k * NC + t];   // coalesced across classes for fixed k
    logits[t] = acc;
  }
  __syncthreads();
  if (t == 0) {
    float m = logits[0];
    for (int c = 1; c < NC; ++c) m = fmaxf(m, logits[c]);
    float sum = 0.0f;
    for (int c = 0; c < NC; ++c) sum += expf(logits[c] - m);
    red[0] = m;
    red[1] = logf(sum);
  }
  __syncthreads();
  if (t < NC) out[(size_t)node * NC + t] = logits[t] - red[0] - red[1];
}

// Tuple tail: reference returns (log_probs, edge_index); indices < 2^24 are
// exact in fp32.
__global__ void edge_copy_kernel(const int* __restrict__ ei,
                                 float* __restrict__ out, int n) {
  int i = blockIdx.x * blockDim.x + threadIdx.x;
  if (i < n) out[i] = (float)ei[i];
}

// ------------------------------------------------------------------ driver
extern "C" void kernel_launch(void* const* d_in, const int* in_sizes, int n_in,
                              void* d_out, int out_size, void* d_ws,
                              size_t ws_size, hipStream_t stream) {
  const float* x    = (const float*)d_in[0];
  const int*   ei   = (const int*)d_in[1];
  const float* w1l  = (const float*)d_in[2];
  const float* w1r  = (const float*)d_in[3];
  const float* b1   = (const float*)d_in[4];
  const float* w2l  = (const float*)d_in[5];
  const float* w2r  = (const float*)d_in[6];
  const float* b2   = (const float*)d_in[7];
  const float* wout = (const float*)d_in[8];
  const float* bout = (const float*)d_in[9];

  const int n_edges = in_sizes[1] / 2;      // edge_index is [2, E]
  const int* src = ei;
  const int* dst = ei + n_edges;

  float* ws   = (float*)d_ws;
  float* dinv = ws;                                   // [NN]
  float* bufA = ws + NN;                              // [NN*DH] agg (and h2)
  float* bufB = bufA + (size_t)NN * DH;               // [NN*DH] h1
  float* out  = (float*)d_out;

  const int feat_elems = NN * DH;
  const int ZB = 256;

  // degrees -> deg_inv
  zero_kernel<<<(NN + ZB - 1) / ZB, ZB, 0, stream>>>(dinv, NN);
  deg_count_kernel<<<(n_edges + ZB - 1) / ZB, ZB, 0, stream>>>(dst, dinv, n_edges);
  deg_invert_kernel<<<(NN + ZB - 1) / ZB, ZB, 0, stream>>>(dinv, NN);

  const long long sc_threads = (long long)n_edges * 32;
  const int sc_blocks = (int)((sc_threads + ZB - 1) / ZB);

  // layer 1: agg1 = scatter(x); h1 = relu(dinv*agg1 @ w1_l + x @ w1_r + b1)
  zero_kernel<<<(feat_elems + ZB - 1) / ZB, ZB, 0, stream>>>(bufA, feat_elems);
  scatter_add_kernel<<<sc_blocks, ZB, 0, stream>>>(x, src, dst, bufA, n_edges);
  gemm_dual_wmma<<<NN / 16, 256, 0, stream>>>(bufA, dinv, w1l, x, w1r, b1, bufB, 1);

  // layer 2: agg2 = scatter(h1); h2 = dinv*agg2 @ w2_l + h1 @ w2_r + b2
  // (h2 written in-place over agg2; safe per barrier in gemm_dual_wmma)
  zero_kernel<<<(feat_elems + ZB - 1) / ZB, ZB, 0, stream>>>(bufA, feat_elems);
  scatter_add_kernel<<<sc_blocks, ZB, 0, stream>>>(bufB, src, dst, bufA, n_edges);
  gemm_dual_wmma<<<NN / 16, 256, 0, stream>>>(bufA, dinv, w2l, bufB, w2r, b2, bufA, 0);

  // logits + log_softmax
  out_logsoftmax_kernel<<<NN, 64, 0, stream>>>(bufA, wout, bout, out);

  // tuple tail: edge_index
  int tail = out_size - NN * NC;
  if (tail > 0) {
    int n = tail < 2 * n_edges ? tail : 2 * n_edges;
    edge_copy_kernel<<<(n + ZB - 1) / ZB, ZB, 0, stream>>>(ei, out + NN * NC, n);
  }
}